// Detect_67937792688194
// MI455X (gfx1250) — compile-verified
//
#include <hip/hip_runtime.h>
#include <hip/hip_bf16.h>
#include <cstdint>

#define NUM_CLASSES 21
#define NUM_CL_M1   20
#define TOPK        200
#define KCAND       400
#define NMS_TH      0.45f
#define CONF_TH     0.01f
#define PNUM        24564
#define BSZ         16

typedef __attribute__((ext_vector_type(2))) float v2f;
typedef __attribute__((ext_vector_type(8))) float v8f;

#if defined(__gfx1250__) && __has_builtin(__builtin_amdgcn_wmma_f32_16x16x4_f32)
#define USE_WMMA 1
#else
#define USE_WMMA 0
#endif

// order-preserving float->uint transform (monotonic increasing)
__device__ __forceinline__ unsigned ordKey(float f) {
  unsigned u = __float_as_uint(f);
  return (u & 0x80000000u) ? ~u : (u | 0x80000000u);
}
__device__ __forceinline__ float unordKey(unsigned k) {
  unsigned u = (k & 0x80000000u) ? (k ^ 0x80000000u) : ~k;
  return __uint_as_float(u);
}

// ---------------------------------------------------------------- decode ----
__global__ void decode_boxes_kernel(const float4* __restrict__ loc,
                                    const float4* __restrict__ prior,
                                    float4* __restrict__ boxes) {
  int idx = blockIdx.x * blockDim.x + threadIdx.x;
  if (idx >= BSZ * PNUM) return;
  int p = idx % PNUM;
  float4 l  = loc[idx];
  float4 pr = prior[p];
  float cx = pr.x + l.x * 0.1f * pr.z;
  float cy = pr.y + l.y * 0.1f * pr.w;
  float w  = pr.z * expf(l.z * 0.2f);
  float h  = pr.w * expf(l.w * 0.2f);
  float x1 = cx - 0.5f * w, y1 = cy - 0.5f * h;
  boxes[idx] = make_float4(x1, y1, x1 + w, y1 + h);
}

// ------------------------------------------- per (image,class) topk + NMS ---
__global__ __launch_bounds__(256) void topk_nms_kernel(
    const float* __restrict__ conf, const float4* __restrict__ boxes,
    float* __restrict__ dets, unsigned long long* __restrict__ keysOut) {
  __shared__ unsigned hist[256];
  __shared__ unsigned long long key64[512];   // 48-bit keys: score<<15 | (0x7fff-p)
  __shared__ float4 cbox[KCAND];
  __shared__ float  area[KCAND];
  __shared__ int    rank_[KCAND];
  __shared__ unsigned mask32[KCAND * 13];     // 400 rows x 400 bits (as 26 u16 halfwords)
  __shared__ unsigned long long s_prefix;
  __shared__ unsigned s_remaining, s_counter;

  const int tid = threadIdx.x;
  const int b = blockIdx.x / NUM_CL_M1;
  const int c = blockIdx.x % NUM_CL_M1;       // class id = c+1 (skip background)
  const float* sp = conf + (size_t)b * PNUM * NUM_CLASSES + (c + 1);

  // zero this block's slice of the sort-key buffer (marks empty slots invalid)
  for (int s = tid; s < TOPK; s += 256)
    keysOut[(size_t)(b * NUM_CL_M1 + c) * TOPK + s] = 0ULL;

  if (tid == 0) { s_prefix = 0ULL; s_remaining = KCAND; s_counter = 0u; }
  __syncthreads();

  // ---- exact radix select of the 400th-largest distinct 48-bit key ----
  for (int round = 0; round < 6; ++round) {
    const int shift = 40 - 8 * round;
    hist[tid] = 0u;
    __syncthreads();
    unsigned long long pre = s_prefix;
    for (int p = tid; p < PNUM; p += 256) {
      unsigned long long k =
          ((unsigned long long)ordKey(sp[(size_t)p * NUM_CLASSES]) << 15) |
          (unsigned long long)(0x7FFFu - (unsigned)p);
      if ((k >> (shift + 8)) == pre)
        atomicAdd(&hist[(unsigned)(k >> shift) & 0xFFu], 1u);
    }
    __syncthreads();
    if (tid == 0) {
      unsigned rem = s_remaining, cum = 0;
      int bin = 255;
      for (; bin >= 0; --bin) { cum += hist[bin]; if (cum >= rem) break; }
      if (bin < 0) bin = 0;
      s_remaining = rem - (cum - hist[bin]);
      s_prefix = (pre << 8) | (unsigned)bin;
    }
    __syncthreads();
  }
  const unsigned long long T = s_prefix;      // rank-400 key; keys are distinct

  // ---- collect exactly the 400 keys >= T, pad to 512 ----
  for (int i = tid; i < 512; i += 256) if (i >= KCAND) key64[i] = 0ULL;
  for (int p = tid; p < PNUM; p += 256) {
    unsigned long long k =
        ((unsigned long long)ordKey(sp[(size_t)p * NUM_CLASSES]) << 15) |
        (unsigned long long)(0x7FFFu - (unsigned)p);
    if (k >= T) {
      unsigned pos = atomicAdd(&s_counter, 1u);
      if (pos < KCAND) key64[pos] = k;
    }
  }
  __syncthreads();

  // ---- bitonic sort 512 keys descending (score desc, index asc) ----
  for (unsigned kk = 2; kk <= 512; kk <<= 1) {
    for (unsigned j = kk >> 1; j > 0; j >>= 1) {
      for (unsigned i = tid; i < 512; i += 256) {
        unsigned ixj = i ^ j;
        if (ixj > i) {
          unsigned long long a = key64[i], d = key64[ixj];
          bool up = ((i & kk) == 0);
          if (up ? (a < d) : (a > d)) { key64[i] = d; key64[ixj] = a; }
        }
      }
      __syncthreads();
    }
  }

  // ---- gather candidate boxes, areas; clear suppression mask ----
  for (int i = tid; i < KCAND; i += 256) {
    unsigned long long k = key64[i];
    int p = 0x7FFF - (int)(k & 0x7FFFu);
    float4 bx = boxes[(size_t)b * PNUM + p];
    cbox[i] = bx;
    area[i] = (bx.z - bx.x) * (bx.w - bx.y);
  }
  for (int w = tid; w < KCAND * 13; w += 256) mask32[w] = 0u;
  __syncthreads();

  // ---- pairwise IoU>thresh bitmask in 16x16 tiles; union via WMMA ----
  {
    const int wave = tid >> 5;
    const int lane = tid & 31;
    const int h = lane >> 4;    // C/D layout half: rows m = r + 8h
    const int n = lane & 15;    // column within tile
    unsigned short* m16 = (unsigned short*)mask32;
    for (int t = wave; t < 25 * 25; t += 8) {
      const int ti = t / 25, tj = t % 25;
      float4 bj = cbox[tj * 16 + n];
      float interR[8];
      v8f cacc;
#pragma unroll
      for (int r = 0; r < 8; ++r) {
        float4 bi = cbox[ti * 16 + r + 8 * h];
        float xx1 = fmaxf(bi.x, bj.x), yy1 = fmaxf(bi.y, bj.y);
        float xx2 = fminf(bi.z, bj.z), yy2 = fminf(bi.w, bj.w);
        float iw = fmaxf(xx2 - xx1, 0.f), ih = fmaxf(yy2 - yy1, 0.f);
        interR[r] = iw * ih;
        cacc[r] = -interR[r];
      }
#if USE_WMMA
      // union[m][n] = area_i[m] + area_j[n] - inter[m][n]
      //   A(16x4): col0 = area_row, col1 = 1      (lanes 16-31 carry K=2,3 -> 0)
      //   B(4x16): row0 = 1,        row1 = area_col
      v2f av, bv;
      av[0] = h ? 0.f : area[ti * 16 + n];
      av[1] = h ? 0.f : 1.f;
      bv[0] = h ? 0.f : 1.f;
      bv[1] = h ? 0.f : area[tj * 16 + n];
      v8f un = __builtin_amdgcn_wmma_f32_16x16x4_f32(
          false, av, false, bv, (short)0, cacc, false, false);
#else
      v8f un;
#pragma unroll
      for (int r = 0; r < 8; ++r)
        un[r] = area[ti * 16 + r + 8 * h] + area[tj * 16 + n] - interR[r];
#endif
#pragma unroll
      for (int r = 0; r < 8; ++r) {
        bool bit = interR[r] > NMS_TH * fmaxf(un[r], 1e-9f);
        unsigned bal = __builtin_amdgcn_ballot_w32(bit);
        if (lane == 0) {
          m16[(ti * 16 + r) * 26 + tj]     = (unsigned short)(bal & 0xFFFFu);
          m16[(ti * 16 + r + 8) * 26 + tj] = (unsigned short)(bal >> 16);
        }
      }
    }
  }
  __syncthreads();

  // ---- greedy NMS scan on one wave32 (sequential in score order) ----
  if (tid < 32) {
    const int lane = tid;
    unsigned Kw = 0u;   // kept bitmask word owned by this lane (lanes 0..12)
    int cnt = 0;
    for (int i = 0; i < KCAND; ++i) {
      unsigned mrow = (lane < 13) ? mask32[i * 13 + lane] : 0u;
      bool anyb = __builtin_amdgcn_ballot_w32((Kw & mrow) != 0u) != 0u;
      float sci = unordKey((unsigned)(key64[i] >> 15));
      bool k = (sci > CONF_TH) && !anyb && (cnt < TOPK);
      if (k && lane == (i >> 5)) Kw |= (1u << (i & 31));
      if (lane == 0) rank_[i] = k ? cnt : -1;
      cnt += k ? 1 : 0;
    }
  }
  __syncthreads();

  // ---- compact kept detections into per-class slots + global sort keys ----
  for (int i = tid; i < KCAND; i += 256) {
    int r = rank_[i];
    if (r < 0) continue;
    unsigned long long k = key64[i];
    unsigned ok = (unsigned)(k >> 15);
    float sc = unordKey(ok);
    float4 bx = cbox[i];
    size_t base = ((size_t)(b * NUM_CL_M1 + c) * TOPK + r) * 6;
    dets[base + 0] = bx.x; dets[base + 1] = bx.y;
    dets[base + 2] = bx.z; dets[base + 3] = bx.w;
    dets[base + 4] = sc;   dets[base + 5] = (float)(c + 1);
    unsigned flat = (unsigned)(c * TOPK + r);
    keysOut[(size_t)b * (NUM_CL_M1 * TOPK) + flat] =
        ((unsigned long long)ok << 13) | (unsigned long long)(0x1FFFu - flat);
  }
}

// ---------------------------------------------- per-image merge & sort ------
__global__ __launch_bounds__(512) void finalize_kernel(
    const unsigned long long* __restrict__ keysIn,
    const float* __restrict__ dets, float* __restrict__ out) {
  __shared__ unsigned long long sk[4096];
  const int tid = threadIdx.x;
  const int b = blockIdx.x;
  for (int i = tid; i < 4096; i += 512)
    sk[i] = (i < NUM_CL_M1 * TOPK) ? keysIn[(size_t)b * (NUM_CL_M1 * TOPK) + i]
                                   : 0ULL;
  __syncthreads();
  for (unsigned kk = 2; kk <= 4096; kk <<= 1) {
    for (unsigned j = kk >> 1; j > 0; j >>= 1) {
      for (unsigned i = tid; i < 4096; i += 512) {
        unsigned ixj = i ^ j;
        if (ixj > i) {
          unsigned long long a = sk[i], d = sk[ixj];
          bool up = ((i & kk) == 0);
          if (up ? (a < d) : (a > d)) { sk[i] = d; sk[ixj] = a; }
        }
      }
      __syncthreads();
    }
  }
  const int ROWS = NUM_CLASSES * TOPK;   // 4200 (includes 200 zero pad rows)
  for (int r = tid; r < ROWS; r += 512) {
    float row[6] = {0.f, 0.f, 0.f, 0.f, 0.f, 0.f};
    if (r < 4096) {
      unsigned long long k = sk[r];
      if (k != 0ULL) {
        unsigned flat = 0x1FFFu - (unsigned)(k & 0x1FFFu);
        size_t base = ((size_t)b * (NUM_CL_M1 * TOPK) + flat) * 6;
#pragma unroll
        for (int t = 0; t < 6; ++t) row[t] = dets[base + t];
      }
    }
    size_t ob = ((size_t)b * ROWS + r) * 6;
#pragma unroll
    for (int t = 0; t < 6; ++t) out[ob + t] = row[t];
  }
}

// ---------------------------------------------------------------- launch ----
extern "C" void kernel_launch(void* const* d_in, const int* in_sizes, int n_in,
                              void* d_out, int out_size, void* d_ws,
                              size_t ws_size, hipStream_t stream) {
  const float4* loc   = (const float4*)d_in[0];   // [16,24564,4]
  const float*  conf  = (const float*)d_in[1];    // [16,24564,21]
  const float4* prior = (const float4*)d_in[2];   // [24564,4]
  float* out = (float*)d_out;                     // [16,4200,6]

  // workspace layout
  float4* boxes = (float4*)d_ws;                                    // 6.29 MB
  float*  dets  = (float*)((char*)d_ws +
                           (size_t)BSZ * PNUM * 4 * sizeof(float)); // 1.54 MB
  unsigned long long* keys = (unsigned long long*)(
      (char*)dets + (size_t)BSZ * NUM_CL_M1 * TOPK * 6 * sizeof(float)); // 512 KB

  int tot = BSZ * PNUM;
  decode_boxes_kernel<<<(tot + 255) / 256, 256, 0, stream>>>(loc, prior, boxes);
  topk_nms_kernel<<<BSZ * NUM_CL_M1, 256, 0, stream>>>(conf, boxes, dets, keys);
  finalize_kernel<<<BSZ, 512, 0, stream>>>(keys, dets, out);
}